// EMAQuantizer_8581344657618
// MI455X (gfx1250) — compile-verified
//
#include <hip/hip_runtime.h>

typedef float v2f __attribute__((ext_vector_type(2)));
typedef float v8f __attribute__((ext_vector_type(8)));

#define K_CODES 4096
#define D_DIM   64
#define N_TOK   (16 * 4096)
#define DECAY_F          0.99f
#define ONE_MINUS_DECAY  0.01f
#define EPSF             1e-5f

// ---- workspace layout (float offsets) ----
#define WS_CSQ     0                         // [4096]  ||c||^2
#define WS_CLUSTER 4096                      // [4096]  batch cluster counts
#define WS_EMBED   8192                      // [4096*64] embed_sum scatter
#define WS_N       (WS_EMBED + K_CODES * D_DIM)  // [1] sum(new_cluster_size)
#define WS_TOTAL   (WS_N + 1)

// ---- output layout (float offsets), reference return order ----
#define OUT_Q   0
#define OUT_IDX (N_TOK * D_DIM)              // 4194304
#define OUT_CB  (OUT_IDX + N_TOK)            // 4259840
#define OUT_CS  (OUT_CB + K_CODES * D_DIM)   // 4521984
#define OUT_EMA (OUT_CS + K_CODES)           // 4526080

#define CHUNK      128                        // codes staged per LDS chunk
#define LDS_STRIDE 66                         // 64 + 2 pad floats (bank-conflict free)

// ---------------------------------------------------------------------------
// Kernel 1: zero scatter accumulators + precompute codebook squared norms
// ---------------------------------------------------------------------------
__global__ void vq_init(const float* __restrict__ cb, float* __restrict__ ws) {
  const int tid = blockIdx.x * blockDim.x + threadIdx.x;
  const int nz = WS_TOTAL - WS_CLUSTER;   // cluster + embed + n
  if (tid < nz) ws[WS_CLUSTER + tid] = 0.0f;
  if (tid < K_CODES) {
    const float* r = cb + tid * D_DIM;
    float s = 0.0f;
#pragma unroll
    for (int d = 0; d < D_DIM; ++d) s += r[d] * r[d];
    ws[WS_CSQ + tid] = s;
  }
}

// ---------------------------------------------------------------------------
// Kernel 2: WMMA distance matmul + argmin + gather + EMA scatter
// One wave = 16 rows of x vs all 4096 codes. 8 waves/block share LDS codebook.
// ---------------------------------------------------------------------------
__global__ __launch_bounds__(256) void vq_assign(const float* __restrict__ x,
                                                 const float* __restrict__ cb,
                                                 float* __restrict__ out,
                                                 float* __restrict__ ws) {
  __shared__ float lds_cb[CHUNK * LDS_STRIDE];
  __shared__ float lds_csq[CHUNK];

  const int lane   = threadIdx.x & 31;
  const int wave   = threadIdx.x >> 5;
  const int laneLo = lane & 15;
  const int hi     = lane >> 4;                       // 0: lanes 0-15, 1: 16-31
  const int rowBase = (blockIdx.x * 8 + wave) * 16;
  const int myRow   = rowBase + laneLo;

  // A fragments: 16x4 f32 tiles covering K=0..63. Lane holds row M=laneLo;
  // VGPR0 = K, VGPR1 = K+1 (lo half) / K+2, K+3 (hi half).
  v2f a[16];
  {
    const float* xr = x + (size_t)myRow * D_DIM;
#pragma unroll
    for (int c = 0; c < 16; ++c)
      a[c] = *(const v2f*)(xr + 4 * c + 2 * hi);
  }

  float minS[8];
  int   minI[8];
#pragma unroll
  for (int r = 0; r < 8; ++r) { minS[r] = 3.4e38f; minI[r] = 0; }

  const float* csq = ws + WS_CSQ;

  for (int chunk = 0; chunk < K_CODES / CHUNK; ++chunk) {
    const int base = chunk * CHUNK;
    __syncthreads();                                  // protect previous chunk
    // ---- stage 128 codes (32 KB) into padded LDS ----
    {
      const float* src = cb + (size_t)base * D_DIM;
#pragma unroll
      for (int i = 0; i < 8; ++i) {
        const int lin = threadIdx.x + 256 * i;        // float4 index, 2048 total
        const int row = lin >> 4;
        const int col = (lin & 15) * 4;
        const float4 v = *(const float4*)(src + row * D_DIM + col);
        float* dst = lds_cb + row * LDS_STRIDE + col; // only 8B-aligned
        ((v2f*)dst)[0] = v2f{v.x, v.y};
        ((v2f*)(dst + 2))[0] = v2f{v.z, v.w};
      }
      if (threadIdx.x < CHUNK) lds_csq[threadIdx.x] = csq[base + threadIdx.x];
    }
    __syncthreads();

    // ---- 8 code tiles of 16: chained f32 WMMA over K=64 ----
#pragma unroll
    for (int t = 0; t < CHUNK / 16; ++t) {
      const float* crow = lds_cb + (t * 16 + laneLo) * LDS_STRIDE;
      v8f acc = {0.f, 0.f, 0.f, 0.f, 0.f, 0.f, 0.f, 0.f};
#pragma unroll
      for (int c = 0; c < 16; ++c) {
        const v2f b = *(const v2f*)(crow + 4 * c + 2 * hi);
        acc = __builtin_amdgcn_wmma_f32_16x16x4_f32(
            false, a[c], false, b, (short)0, acc, false, false);
      }
      const int   code = base + t * 16 + laneLo;      // lane's code column
      const float cs   = lds_csq[t * 16 + laneLo];
#pragma unroll
      for (int r = 0; r < 8; ++r) {
        const float s = cs - 2.0f * acc[r];           // ||c||^2 - 2 x.c
        if (s < minS[r]) { minS[r] = s; minI[r] = code; }
      }
    }
  }

  // ---- argmin reduction across the 16 lanes of each half (rows M=r / r+8) ----
#pragma unroll
  for (int r = 0; r < 8; ++r) {
    float s = minS[r];
    int   i = minI[r];
#pragma unroll
    for (int m = 1; m < 16; m <<= 1) {
      const float os = __shfl_xor(s, m, 32);
      const int   oi = __shfl_xor(i, m, 32);
      if (os < s || (os == s && oi < i)) { s = os; i = oi; }  // first-occurrence
    }
    minS[r] = s;
    minI[r] = i;
  }

  // indices out (reference dtype coerced to float by harness)
  if (laneLo == 0) {
#pragma unroll
    for (int r = 0; r < 8; ++r)
      out[OUT_IDX + rowBase + hi * 8 + r] = (float)minI[r];
  }

  float* q          = out + OUT_Q;
  float* cluster_ws = ws + WS_CLUSTER;
  float* embed_ws   = ws + WS_EMBED;

  // gather quantized rows + scatter EMA statistics
  for (int row = 0; row < 16; ++row) {
    const int srcLane = (row < 8) ? 0 : 16;
    const int id = __shfl(minI[row & 7], srcLane, 32);
    // quantized[row] = codebook[id]  (float2 per lane)
    const v2f cv = *(const v2f*)(cb + (size_t)id * D_DIM + lane * 2);
    *(v2f*)(q + (size_t)(rowBase + row) * D_DIM + lane * 2) = cv;
    // embed_sum[id] += x[row]
    const v2f xv = *(const v2f*)(x + (size_t)(rowBase + row) * D_DIM + lane * 2);
    atomicAdd(&embed_ws[(size_t)id * D_DIM + lane * 2 + 0], xv.x);
    atomicAdd(&embed_ws[(size_t)id * D_DIM + lane * 2 + 1], xv.y);
    if (lane == row) atomicAdd(&cluster_ws[id], 1.0f);  // batch_cluster[id] += 1
  }
}

// ---------------------------------------------------------------------------
// Kernel 3: EMA updates + total count n
// ---------------------------------------------------------------------------
__global__ void vq_ema(const float* __restrict__ cluster_size,
                       const float* __restrict__ ema_w,
                       float* __restrict__ out, float* __restrict__ ws) {
  const int i = blockIdx.x * blockDim.x + threadIdx.x;
  if (i >= K_CODES * D_DIM) return;
  out[OUT_EMA + i] = DECAY_F * ema_w[i] + ONE_MINUS_DECAY * ws[WS_EMBED + i];
  if ((i & 63) == 0) {
    const int k = i >> 6;
    const float ncs = DECAY_F * cluster_size[k] + ONE_MINUS_DECAY * ws[WS_CLUSTER + k];
    out[OUT_CS + k] = ncs;
    atomicAdd(&ws[WS_N], ncs);
  }
}

// ---------------------------------------------------------------------------
// Kernel 4: Laplace-smoothed normalization -> new codebook
// ---------------------------------------------------------------------------
__global__ void vq_codebook(float* __restrict__ out, const float* __restrict__ ws) {
  const int i = blockIdx.x * blockDim.x + threadIdx.x;
  if (i >= K_CODES * D_DIM) return;
  const int k = i >> 6;
  const float n = ws[WS_N];
  const float ncs = out[OUT_CS + k];
  const float smoothed = (ncs + EPSF) / (n + K_CODES * EPSF) * n;
  out[OUT_CB + i] = out[OUT_EMA + i] / smoothed;
}

extern "C" void kernel_launch(void* const* d_in, const int* in_sizes, int n_in,
                              void* d_out, int out_size, void* d_ws, size_t ws_size,
                              hipStream_t stream) {
  (void)in_sizes; (void)n_in; (void)out_size; (void)ws_size;
  const float* x            = (const float*)d_in[0];
  const float* cb           = (const float*)d_in[1];
  const float* cluster_size = (const float*)d_in[2];
  const float* ema_w        = (const float*)d_in[3];
  float* out = (float*)d_out;
  float* ws  = (float*)d_ws;

  const int initThreads = WS_TOTAL - WS_CLUSTER;          // 266241 zeros (+csq)
  vq_init<<<(initThreads + 255) / 256, 256, 0, stream>>>(cb, ws);
  vq_assign<<<N_TOK / 128, 256, 0, stream>>>(x, cb, out, ws);   // 512 blocks
  vq_ema<<<(K_CODES * D_DIM + 255) / 256, 256, 0, stream>>>(cluster_size, ema_w, out, ws);
  vq_codebook<<<(K_CODES * D_DIM + 255) / 256, 256, 0, stream>>>(out, ws);
}